// GNN_57973468562126
// MI455X (gfx1250) — compile-verified
//
#include <hip/hip_runtime.h>

// ---------------- problem constants ----------------
#define NN    3686               // nodes
#define NPAD  3696               // 231 * 16 (row padding for 16x16 tiles)
#define MT    231                // NPAD / 16
#define NB    8                  // batch
#define HID   64
#define CW    512                // NB * HID  (Z layout: [NPAD rows, 512 cols], col = b*64+h)
#define OH    128                // OUT_HID
#define OD    10                 // OUT_DIM
#define KTOT  (NN * HID)         // 235904  readout-K
#define SPLITK 128
#define KC    1844               // multiple of 4, SPLITK*KC = 236032 >= KTOT

typedef __attribute__((ext_vector_type(2))) float v2f;
typedef __attribute__((ext_vector_type(8))) float v8f;

__device__ __forceinline__ v8f wmma_f32(v2f a, v2f b, v8f c) {
  // V_WMMA_F32_16X16X4_F32 : D = A(16x4) * B(4x16) + C(16x16), full fp32
  return __builtin_amdgcn_wmma_f32_16x16x4_f32(false, a, false, b, (short)0, c, false, false);
}

// ---------------- deg[n] = 1 + rowsum(adj) ----------------
__global__ void deg_kernel(const float* __restrict__ adj, float* __restrict__ deg) {
  __shared__ float sm[256];
  const int n = blockIdx.x;
  if (n >= NN) { if (threadIdx.x == 0) deg[n] = 1.0f; return; }
  float s = 0.0f;
  for (int m = threadIdx.x; m < NN; m += 256) s += adj[(size_t)n * NN + m];
  sm[threadIdx.x] = s;
  __syncthreads();
  for (int off = 128; off > 0; off >>= 1) {
    if (threadIdx.x < off) sm[threadIdx.x] += sm[threadIdx.x + off];
    __syncthreads();
  }
  if (threadIdx.x == 0) deg[n] = 1.0f + sm[0];
}

// ---------------- Arow = adj @ map_W[1:]  ([NPAD x 64], batch-invariant) ----------------
__global__ void gemm_arow(const float* __restrict__ adj,
                          const float* __restrict__ mapW,
                          float* __restrict__ Arow) {
  const int lane = threadIdx.x;
  const int l15  = lane & 15;
  const int m0   = blockIdx.x * 16;
  const int mr   = m0 + l15;
  const bool mrOK = mr < NN;
  const size_t aoff = (size_t)(mrOK ? mr : 0) * NN;     // clamped, always in-bounds
  const int kb = (lane < 16) ? 0 : 2;

  v8f acc[4] = {};                                      // 4 col tiles -> all 64 cols
  for (int k = 0; k < 3684; k += 4) {                   // main loop: k0,k1 < NN always
    const int k0 = k + kb;
    float a0 = adj[aoff + k0];
    float a1 = adj[aoff + k0 + 1];
    v2f a; a.x = mrOK ? a0 : 0.0f; a.y = mrOK ? a1 : 0.0f;
#pragma unroll
    for (int t = 0; t < 4; ++t) {
      v2f bb;
      bb.x = mapW[(size_t)(1 + k0) * 64 + 16 * t + l15];
      bb.y = mapW[(size_t)(2 + k0) * 64 + 16 * t + l15];
      acc[t] = wmma_f32(a, bb, acc[t]);
    }
  }
  { // tail k = 3684 : guard K against NN
    const int k0 = 3684 + kb, k1 = k0 + 1;
    const bool ok0 = k0 < NN, ok1 = k1 < NN;
    const int k0c = ok0 ? k0 : 0, k1c = ok1 ? k1 : 0;
    float a0 = adj[aoff + k0c];
    float a1 = adj[aoff + k1c];
    v2f a; a.x = (mrOK && ok0) ? a0 : 0.0f; a.y = (mrOK && ok1) ? a1 : 0.0f;
#pragma unroll
    for (int t = 0; t < 4; ++t) {
      float b0 = mapW[(size_t)(1 + k0c) * 64 + 16 * t + l15];
      float b1 = mapW[(size_t)(1 + k1c) * 64 + 16 * t + l15];
      v2f bb; bb.x = ok0 ? b0 : 0.0f; bb.y = ok1 ? b1 : 0.0f;
      acc[t] = wmma_f32(a, bb, acc[t]);
    }
  }
  const int mbase = m0 + ((lane < 16) ? 0 : 8);
#pragma unroll
  for (int t = 0; t < 4; ++t)
#pragma unroll
    for (int r = 0; r < 8; ++r)
      Arow[(size_t)(mbase + r) * 64 + 16 * t + l15] = acc[t][r];
}

// ---------------- Z0[n][b*64+d] = x[b,n]*map_W[0][d] + Arow[n][d] + map_b[d] ----------------
__global__ void init_z(const float* __restrict__ x, const float* __restrict__ mapW,
                       const float* __restrict__ mapb, const float* __restrict__ Arow,
                       float* __restrict__ Z) {
  const int idx = blockIdx.x * 256 + threadIdx.x;
  if (idx >= NPAD * CW) return;
  const int n = idx >> 9;
  const int c = idx & 511;
  const int b = c >> 6;
  const int d = c & 63;
  float v = 0.0f;                                        // pad rows stay exactly zero
  if (n < NN) v = x[b * NN + n] * mapW[d] + Arow[n * 64 + d] + mapb[d];
  Z[idx] = v;
}

// ---------------- Hh = Z @ W (per-batch block-diagonal feature GEMM) ----------------
__global__ void gemm_feat(const float* __restrict__ Z, const float* __restrict__ W,
                          float* __restrict__ Hh) {
  const int lane = threadIdx.x;
  const int l15  = lane & 15;
  const int m0   = blockIdx.x * 16;
  const int b    = blockIdx.y;                           // batch
  const int kb   = (lane < 16) ? 0 : 2;
  v8f acc[4] = {};
#pragma unroll
  for (int k = 0; k < HID; k += 4) {
    v2f a;
    a.x = Z[(size_t)(m0 + l15) * CW + b * HID + k + kb];
    a.y = Z[(size_t)(m0 + l15) * CW + b * HID + k + kb + 1];
#pragma unroll
    for (int t = 0; t < 4; ++t) {
      v2f bb;
      bb.x = W[(k + kb) * HID + 16 * t + l15];
      bb.y = W[(k + kb + 1) * HID + 16 * t + l15];
      acc[t] = wmma_f32(a, bb, acc[t]);
    }
  }
  const int mbase = m0 + ((lane < 16) ? 0 : 8);
#pragma unroll
  for (int t = 0; t < 4; ++t)
#pragma unroll
    for (int r = 0; r < 8; ++r)
      Hh[(size_t)(mbase + r) * CW + b * HID + 16 * t + l15] = acc[t][r];
}

// ---------------- Zn = (adj @ Hh + Hh) / deg + bias (fused epilogue) ----------------
__global__ void gemm_prop(const float* __restrict__ adj, const float* __restrict__ Hh,
                          const float* __restrict__ deg, const float* __restrict__ bias,
                          float* __restrict__ Zn) {
  const int lane = threadIdx.x;
  const int l15  = lane & 15;
  const int m0   = blockIdx.x * 16;
  const int c0   = blockIdx.y * 64;                      // one batch's 64 columns
  const int mr   = m0 + l15;
  const bool mrOK = mr < NN;
  const size_t aoff = (size_t)(mrOK ? mr : 0) * NN;
  const int kb = (lane < 16) ? 0 : 2;

  v8f acc[4] = {};
  for (int k = 0; k < 3684; k += 4) {
    const int k0 = k + kb;
    float a0 = adj[aoff + k0];
    float a1 = adj[aoff + k0 + 1];
    v2f a; a.x = mrOK ? a0 : 0.0f; a.y = mrOK ? a1 : 0.0f;
#pragma unroll
    for (int t = 0; t < 4; ++t) {
      v2f bb;
      bb.x = Hh[(size_t)k0 * CW + c0 + 16 * t + l15];
      bb.y = Hh[(size_t)(k0 + 1) * CW + c0 + 16 * t + l15];
      acc[t] = wmma_f32(a, bb, acc[t]);
    }
  }
  { // tail k = 3684 ; Hh rows up to 3687 < NPAD are in-bounds and zero-padded
    const int k0 = 3684 + kb, k1 = k0 + 1;
    const bool ok0 = k0 < NN, ok1 = k1 < NN;
    const int k0c = ok0 ? k0 : 0, k1c = ok1 ? k1 : 0;
    float a0 = adj[aoff + k0c];
    float a1 = adj[aoff + k1c];
    v2f a; a.x = (mrOK && ok0) ? a0 : 0.0f; a.y = (mrOK && ok1) ? a1 : 0.0f;
#pragma unroll
    for (int t = 0; t < 4; ++t) {
      v2f bb;
      bb.x = Hh[(size_t)k0 * CW + c0 + 16 * t + l15];
      bb.y = Hh[(size_t)k1 * CW + c0 + 16 * t + l15];
      acc[t] = wmma_f32(a, bb, acc[t]);
    }
  }

  const int mbase = m0 + ((lane < 16) ? 0 : 8);
#pragma unroll
  for (int t = 0; t < 4; ++t) {
    const int col = c0 + 16 * t + l15;
    const float bj = bias[16 * t + l15];                 // col % 64
#pragma unroll
    for (int r = 0; r < 8; ++r) {
      const int m = mbase + r;
      float v = 0.0f;                                    // keep pad rows exactly zero
      if (m < NN) v = (acc[t][r] + Hh[(size_t)m * CW + col]) / deg[m] + bj;
      Zn[(size_t)m * CW + col] = v;
    }
  }
}

// ---------------- out1 = zeros, then split-K readout GEMM [8 x 235904]@[235904 x 128] ----------------
__global__ void zero_f32(float* __restrict__ p, int n) {
  const int i = blockIdx.x * 256 + threadIdx.x;
  if (i < n) p[i] = 0.0f;
}

__global__ void gemm_r1(const float* __restrict__ Z, const float* __restrict__ r1W,
                        float* __restrict__ out1) {
  const int lane = threadIdx.x;
  const int l15  = lane & 15;
  const int o0   = blockIdx.x * 16;                      // column tile of OUT_HID
  const int kstart = blockIdx.y * KC;                    // split-K chunk
  const int kb   = (lane < 16) ? 0 : 2;
  const bool bOK = l15 < 8;                              // A rows 8..15 are padding
  const int bcol = l15 * 64;

  v8f acc = {};
  for (int kk = 0; kk < KC; kk += 4) {
    const int k0 = kstart + kk + kb, k1 = k0 + 1;
    const bool ok0 = k0 < KTOT, ok1 = k1 < KTOT;
    const int k0c = ok0 ? k0 : 0, k1c = ok1 ? k1 : 0;
    // A[b][k] = Z[k/64][b*64 + k%64]  (Z pad rows are zero)
    float a0 = Z[(size_t)(k0c >> 6) * CW + bcol + (k0c & 63)];
    float a1 = Z[(size_t)(k1c >> 6) * CW + bcol + (k1c & 63)];
    v2f a; a.x = (bOK && ok0) ? a0 : 0.0f; a.y = (bOK && ok1) ? a1 : 0.0f;
    float b0 = r1W[(size_t)k0c * OH + o0 + l15];
    float b1 = r1W[(size_t)k1c * OH + o0 + l15];
    v2f b; b.x = ok0 ? b0 : 0.0f; b.y = ok1 ? b1 : 0.0f;
    acc = wmma_f32(a, b, acc);
  }
  const int mbase = (lane < 16) ? 0 : 8;
#pragma unroll
  for (int r = 0; r < 8; ++r) {
    const int m = mbase + r;
    if (m < NB) atomicAdd(&out1[m * OH + o0 + l15], acc[r]);
  }
}

// ---------------- BN(eval) + ReLU + [8x128]@[128x10] + biases ----------------
__global__ void readout2(const float* __restrict__ out1, const float* __restrict__ r1b,
                         const float* __restrict__ g, const float* __restrict__ be,
                         const float* __restrict__ mu, const float* __restrict__ var,
                         const float* __restrict__ r2W, const float* __restrict__ r2b,
                         float* __restrict__ out) {
  const int t = blockIdx.x * blockDim.x + threadIdx.x;
  if (t >= NB * OD) return;
  const int bi = t / OD, o = t % OD;
  float s = 0.0f;
  for (int j = 0; j < OH; ++j) {
    float h = out1[bi * OH + j] + r1b[j];
    h = (h - mu[j]) * rsqrtf(var[j] + 1e-5f) * g[j] + be[j];
    h = fmaxf(h, 0.0f);
    s += h * r2W[j * OD + o];
  }
  out[t] = s + r2b[o];
}

// ---------------- host launch ----------------
extern "C" void kernel_launch(void* const* d_in, const int* in_sizes, int n_in,
                              void* d_out, int out_size, void* d_ws, size_t ws_size,
                              hipStream_t stream) {
  const float* x    = (const float*)d_in[0];
  const float* adj  = (const float*)d_in[1];
  const float* mapW = (const float*)d_in[2];
  const float* mapb = (const float*)d_in[3];
  const float* cW[3] = {(const float*)d_in[4], (const float*)d_in[6], (const float*)d_in[8]};
  const float* cb[3] = {(const float*)d_in[5], (const float*)d_in[7], (const float*)d_in[9]};
  const float* r1W  = (const float*)d_in[10];
  const float* r1b  = (const float*)d_in[11];
  const float* bng  = (const float*)d_in[12];
  const float* bnb  = (const float*)d_in[13];
  const float* bnm  = (const float*)d_in[14];
  const float* bnv  = (const float*)d_in[15];
  const float* r2W  = (const float*)d_in[16];
  const float* r2b  = (const float*)d_in[17];

  // workspace carve (fp32): deg | Arow | Z0 | Z1 | Hh | out1  (~23.7 MB total)
  float* ws   = (float*)d_ws;
  float* deg  = ws;
  float* Arow = deg + NPAD;
  float* Z0   = Arow + (size_t)NPAD * 64;
  float* Z1   = Z0 + (size_t)NPAD * CW;
  float* Hh   = Z1 + (size_t)NPAD * CW;
  float* out1 = Hh + (size_t)NPAD * CW;

  deg_kernel<<<NPAD, 256, 0, stream>>>(adj, deg);
  gemm_arow<<<MT, 32, 0, stream>>>(adj, mapW, Arow);
  init_z<<<(NPAD * CW + 255) / 256, 256, 0, stream>>>(x, mapW, mapb, Arow, Z0);

  float* zin = Z0; float* zout = Z1;
  for (int l = 0; l < 3; ++l) {
    gemm_feat<<<dim3(MT, NB), 32, 0, stream>>>(zin, cW[l], Hh);
    gemm_prop<<<dim3(MT, NB), 32, 0, stream>>>(adj, Hh, deg, cb[l], zout);
    float* tmp = zin; zin = zout; zout = tmp;
  }

  zero_f32<<<(NB * OH + 255) / 256, 256, 0, stream>>>(out1, NB * OH);
  gemm_r1<<<dim3(OH / 16, SPLITK), 32, 0, stream>>>(zin, r1W, out1);
  readout2<<<1, 128, 0, stream>>>(out1, r1b, bng, bnb, bnm, bnv, r2W, r2b, (float*)d_out);
}